// ConjGradLoss_PlainNorm_82978768159395
// MI455X (gfx1250) — compile-verified
//
#include <hip/hip_runtime.h>
#include <hip/hip_bf16.h>

// ---------------------------------------------------------------------------
// ConjGradLoss (block SpMV + per-segment CG alpha + relative L2 loss)
// MI455X / gfx1250, wave32.
//
// Roofline: ~420 MB streamed (mv 134MB + idx 134MB + mask 8MB + gathers +
// atomics) -> ~18us floor @ 23.3 TB/s. FLOPs negligible -> pure bandwidth.
// NT hints on single-use streams keep the 2MB d / 2MB Ad vectors L2-resident.
// Scatter uses native global_atomic_add_f32 (inline asm, no CAS loop).
// Reductions use V_WMMA_F32_16X16X4_F32 (full f32 precision) on the matrix
// pipe + one ds_swizzle SWAPX16 fold.
// ---------------------------------------------------------------------------

typedef __attribute__((ext_vector_type(2))) float fv2;
typedef __attribute__((ext_vector_type(4))) float fv4;
typedef __attribute__((ext_vector_type(8))) float fv8;

#define N_NODES 262144
#define N_SEGS  64
#define PER_SEG 4096
#define N_EDGES 8388608
#define EPS_F   1e-6f

// ---------------------------------------------------------------------------
// Native f32 atomic add, device scope, no return value.
// Guaranteed to emit GLOBAL_ATOMIC_ADD_F32 (STOREcnt-tracked); avoids any
// possible CAS-loop expansion of the generic atomic builtin. Target memory is
// coarse-grained HBM, so the hardware fadd is safe.
// ---------------------------------------------------------------------------
__device__ __forceinline__ void atomic_add_f32_dev(float* addr, float v) {
    asm volatile("global_atomic_add_f32 %0, %1, off"
                 :
                 : "v"(addr), "v"(v)
                 : "memory");
}

// ---------------------------------------------------------------------------
// Exact f32 wave(32) sum via V_WMMA_F32_16X16X4_F32.
// A (16x4): lane m (0-15) supplies K0=v_m, K1=0; lane m+16 supplies K2=v_{m+16},
// K3=0 (documented 32-bit A layout). B = all-ones (layout independent).
// => D[m][n] = v_m + v_{m+16} for all n. A lane's 8 D VGPRs hold rows
// {r, r+8} -> per-lane sum covers one half of the wave; one ds_swizzle
// SWAPX16 (offset 0x401F: xor=0x10, and=0x1F) completes the exact 32-lane
// sum. All lanes return the total.
// EXEC must be all 1s at call sites (no divergence in callers).
// ---------------------------------------------------------------------------
__device__ __forceinline__ float wave_sum32(float v) {
    fv2 a; a[0] = v;    a[1] = 0.0f;
    fv2 b; b[0] = 1.0f; b[1] = 1.0f;
    fv8 c = {0.f, 0.f, 0.f, 0.f, 0.f, 0.f, 0.f, 0.f};
    c = __builtin_amdgcn_wmma_f32_16x16x4_f32(
            /*neg_a=*/false, a, /*neg_b=*/false, b,
            /*c_mod=*/(short)0, c, /*reuse_a=*/false, /*reuse_b=*/false);
    float s = ((c[0] + c[1]) + (c[2] + c[3])) + ((c[4] + c[5]) + (c[6] + c[7]));
    int sw = __builtin_amdgcn_ds_swizzle(__float_as_int(s), 0x401F); // SWAPX16
    s += __int_as_float(sw);
    return s;
}

// ---------------------------------------------------------------------------
// Kernel 1: zero Ad accumulator (in workspace) and the output scalar.
// ---------------------------------------------------------------------------
__global__ void cgl_zero(float* __restrict__ Ad, float* __restrict__ out) {
    int i = blockIdx.x * 256 + threadIdx.x;
    if (i < N_NODES * 2) Ad[i] = 0.0f;
    if (i == 0) out[0] = 0.0f;
}

// ---------------------------------------------------------------------------
// Kernel 2: masked block SpMV, one edge per thread.
//   Ad[row] += A_e(2x2) @ d[col]
// Streams (mv, rows, cols, mask) are single-use -> non-temporal loads.
// Gather of d[col] uses default caching (d is 2MB, L2 resident).
// ---------------------------------------------------------------------------
__global__ void cgl_spmv(const fv4* __restrict__ mv,
                         const float2* __restrict__ dvec,
                         const long long* __restrict__ rows,
                         const long long* __restrict__ cols,
                         const unsigned char* __restrict__ mask,
                         float* __restrict__ Ad) {
    int e = blockIdx.x * 256 + threadIdx.x;   // grid sized exactly E/256

    fv4           m  = __builtin_nontemporal_load(mv + e);
    long long     r  = __builtin_nontemporal_load(rows + e);
    long long     cc = __builtin_nontemporal_load(cols + e);
    unsigned char mk = __builtin_nontemporal_load(mask + e);

    if (mk) {
        float2 dv = dvec[cc];                 // L2-resident gather
        // contrib[i] = sum_j mv[i][j] * d[col][j]
        float c0 = m[0] * dv.x + m[1] * dv.y;
        float c1 = m[2] * dv.x + m[3] * dv.y;
        float* dst = Ad + ((size_t)r * 2);
        atomic_add_f32_dev(dst + 0, c0);
        atomic_add_f32_dev(dst + 1, c1);
    }
}

// ---------------------------------------------------------------------------
// Kernel 3: one 256-thread block (8 waves) per segment of 4096 nodes.
// Phase A: rd = sum r.d ; dq = sum d.Ad  -> alpha = rd/(dq+eps)
// Phase B: err2 = sum |alpha*d - gt|^2 ; gt2 = sum |gt|^2
// loss += (err2/(gt2+eps)) / N_SEGS   (atomic over 64 blocks)
// All WMMA reductions run with full EXEC (no divergence before them).
// ---------------------------------------------------------------------------
__global__ void cgl_loss(const float2* __restrict__ d,
                         const float2* __restrict__ residual,
                         const float2* __restrict__ gt,
                         const float2* __restrict__ Ad,
                         float* __restrict__ out) {
    const int seg  = blockIdx.x;
    const int tid  = threadIdx.x;
    const int base = seg * PER_SEG;
    const int wave = tid >> 5;

    __shared__ float s_a[8];
    __shared__ float s_b[8];
    __shared__ float s_alpha;

    // ---- Phase A: per-thread partials over 16 strided nodes ----
    float rd = 0.0f, dq = 0.0f;
#pragma unroll 4
    for (int i = tid; i < PER_SEG; i += 256) {
        int n = base + i;
        float2 dv = d[n];
        float2 rv = residual[n];
        float2 av = Ad[n];
        rd += rv.x * dv.x + rv.y * dv.y;
        dq += dv.x * av.x + dv.y * av.y;
    }
    float rdw = wave_sum32(rd);
    float dqw = wave_sum32(dq);
    if ((tid & 31) == 0) { s_a[wave] = rdw; s_b[wave] = dqw; }
    __syncthreads();
    if (tid == 0) {
        float RD = 0.0f, DQ = 0.0f;
#pragma unroll
        for (int w = 0; w < 8; ++w) { RD += s_a[w]; DQ += s_b[w]; }
        s_alpha = RD / (DQ + EPS_F);
    }
    __syncthreads();
    const float alpha = s_alpha;

    // ---- Phase B: relative L2 partials ----
    float e2 = 0.0f, g2 = 0.0f;
#pragma unroll 4
    for (int i = tid; i < PER_SEG; i += 256) {
        int n = base + i;
        float2 dv = d[n];
        float2 gv = gt[n];
        float ex = alpha * dv.x - gv.x;
        float ey = alpha * dv.y - gv.y;
        e2 += ex * ex + ey * ey;
        g2 += gv.x * gv.x + gv.y * gv.y;
    }
    float e2w = wave_sum32(e2);
    float g2w = wave_sum32(g2);
    if ((tid & 31) == 0) { s_a[wave] = e2w; s_b[wave] = g2w; }
    __syncthreads();
    if (tid == 0) {
        float E2 = 0.0f, G2 = 0.0f;
#pragma unroll
        for (int w = 0; w < 8; ++w) { E2 += s_a[w]; G2 += s_b[w]; }
        float contrib = (E2 / (G2 + EPS_F)) * (1.0f / (float)N_SEGS);
        atomic_add_f32_dev(out, contrib);
    }
}

// ---------------------------------------------------------------------------
// Inputs (setup_inputs order):
//   0 d[N,2] f32, 1 residual[N,2] f32, 2 gt[N,2] f32,
//   3 matrix_values[E,2,2] f32, 4 L_values (UNUSED), 5 edge_index[2,E] i64,
//   6 mask[E] bool(u8), 7 batch_ids (UNUSED: segments are i/4096 by spec).
// Output: single f32 scalar. Workspace: Ad accumulator (N*2 f32 = 2MB).
// ---------------------------------------------------------------------------
extern "C" void kernel_launch(void* const* d_in, const int* in_sizes, int n_in,
                              void* d_out, int out_size, void* d_ws, size_t ws_size,
                              hipStream_t stream) {
    const float*         dvec  = (const float*)d_in[0];
    const float*         resid = (const float*)d_in[1];
    const float*         gt    = (const float*)d_in[2];
    const float*         mv    = (const float*)d_in[3];
    const long long*     eidx  = (const long long*)d_in[5];
    const unsigned char* mask  = (const unsigned char*)d_in[6];

    float* out = (float*)d_out;
    float* Ad  = (float*)d_ws;              // N_NODES*2 floats

    const long long* rows = eidx;           // edge_index[0, :]
    const long long* cols = eidx + N_EDGES; // edge_index[1, :]

    cgl_zero<<<(N_NODES * 2 + 255) / 256, 256, 0, stream>>>(Ad, out);

    cgl_spmv<<<N_EDGES / 256, 256, 0, stream>>>(
        (const fv4*)mv, (const float2*)dvec, rows, cols, mask, Ad);

    cgl_loss<<<N_SEGS, 256, 0, stream>>>(
        (const float2*)dvec, (const float2*)resid, (const float2*)gt,
        (const float2*)Ad, out);
}